// ClosestPointLoss_69054484185799
// MI455X (gfx1250) — compile-verified
//
#include <hip/hip_runtime.h>

typedef float v2f __attribute__((ext_vector_type(2)));
typedef float v8f __attribute__((ext_vector_type(8)));

#define N_OUT   131072
#define M_TGT   16384
#define CHUNK   2048            // targets per LDS buffer (32 KB each, double buffered)
#define NCHUNK  (M_TGT / CHUNK) // 8
#define BLOCK   256             // 8 waves (wave32)
#define WAVES   8
#define LPT     (CHUNK / BLOCK) // float4 elements staged per thread per chunk = 8
#define ROWS_PER_WAVE 32        // two 16-row A tiles per wave
#define ROWS_PER_WG   (WAVES * ROWS_PER_WAVE)   // 256
#define N_WGS   (N_OUT / ROWS_PER_WG)           // 512

// Prep: t4[j] = (x, y, z, x^2+y^2+z^2) so B columns carry ||b||^2 in K=3.
__global__ void prep_targets(const float* __restrict__ tgt, float4* __restrict__ t4) {
    int j = blockIdx.x * blockDim.x + threadIdx.x;
    if (j < M_TGT) {
        float x = tgt[j * 3 + 0], y = tgt[j * 3 + 1], z = tgt[j * 3 + 2];
        t4[j] = make_float4(x, y, z, x * x + y * y + z * z);
    }
}

// Async DMA of one chunk of float4 targets into an LDS buffer (GVS mode:
// vdst = per-lane LDS byte offset, vaddr = per-lane byte offset, saddr = base).
// Tracked by ASYNCcnt; no VGPR data registers are consumed.
__device__ __forceinline__ void stage_chunk_async(const float4* __restrict__ gbase,
                                                  int chunk_elem_base,
                                                  float4* sdst, int tid) {
#pragma unroll
    for (int i = 0; i < LPT; i++) {
        const int idx = tid + i * BLOCK;
        uint32_t ldsoff = (uint32_t)(uintptr_t)(&sdst[idx]);          // LDS byte offset
        uint32_t voff   = (uint32_t)(chunk_elem_base + idx) * 16u;    // global byte offset
        asm volatile("global_load_async_to_lds_b128 %0, %1, %2"
                     :: "v"(ldsoff), "v"(voff), "s"(gbase) : "memory");
    }
}

__device__ __forceinline__ void wait_async_done() {
    asm volatile("s_wait_asynccnt 0x0" ::: "memory");
}

__global__ void __launch_bounds__(BLOCK)
cpl_main(const float* __restrict__ pts, const float4* __restrict__ t4,
         float* __restrict__ partials) {
    __shared__ float4 sbuf[2][CHUNK];   // double-buffered target chunks
    __shared__ float  wsum[WAVES];

    const int tid  = threadIdx.x;
    const int lane = tid & 31;
    const int wave = tid >> 5;
    const bool lo  = (lane < 16);

    // Kick off DMA of chunk 0 before doing anything else.
    stage_chunk_async(t4, 0, sbuf[0], tid);

    // Two 16-row A tiles per wave.
    const int base = blockIdx.x * ROWS_PER_WG + wave * ROWS_PER_WAVE;
    const int r0   = base + (lane & 15);
    const int r1   = r0 + 16;

    float x0 = pts[r0 * 3 + 0], y0 = pts[r0 * 3 + 1], z0 = pts[r0 * 3 + 2];
    float x1 = pts[r1 * 3 + 0], y1 = pts[r1 * 3 + 1], z1 = pts[r1 * 3 + 2];
    float norm0 = x0 * x0 + y0 * y0 + z0 * z0;
    float norm1 = x1 * x1 + y1 * y1 + z1 * z1;

    // A layout (16x4 f32): VGPR0 = K0 (lanes 0-15) / K2 (lanes 16-31),
    //                      VGPR1 = K1 (lanes 0-15) / K3 (lanes 16-31).
    v2f a0, a1;
    a0.x = lo ? (-2.0f * x0) : (-2.0f * z0);
    a0.y = lo ? (-2.0f * y0) : 1.0f;
    a1.x = lo ? (-2.0f * x1) : (-2.0f * z1);
    a1.y = lo ? (-2.0f * y1) : 1.0f;

    v8f m0, m1;
#pragma unroll
    for (int i = 0; i < 8; i++) { m0[i] = 3.4e38f; m1[i] = 3.4e38f; }

    const int lsel = lo ? 0 : 2;   // lanes 0-15 read (x,y)=K0,K1; lanes 16-31 read (z,n2)=K2,K3

    wait_async_done();
    __syncthreads();               // chunk 0 resident in sbuf[0]

    int cur = 0;
    for (int c = 0; c < NCHUNK; c++) {
        const bool have_next = (c + 1) < NCHUNK;
        // DMA the next chunk into the idle buffer; runs under the WMMA loop.
        // Safe: every wave passed the previous barrier, so nobody still reads it.
        if (have_next)
            stage_chunk_async(t4, (c + 1) * CHUNK, sbuf[cur ^ 1], tid);

        const float* sflat = (const float*)sbuf[cur];
#pragma unroll 4
        for (int t = 0; t < CHUNK / 16; t++) {
            const int tl = t * 16 + (lane & 15);
            const float2 b2 = *(const float2*)&sflat[tl * 4 + lsel];  // ds_load_b64, conflict-free
            v2f b; b.x = b2.x; b.y = b2.y;
            v8f cz = {};
            // D = ||b||^2 - 2 a.b per 16x16 tile
            v8f d0 = __builtin_amdgcn_wmma_f32_16x16x4_f32(
                false, a0, false, b, (short)0, cz, false, false);
            v8f d1 = __builtin_amdgcn_wmma_f32_16x16x4_f32(
                false, a1, false, b, (short)0, cz, false, false);
#pragma unroll
            for (int i = 0; i < 8; i++) {
                m0[i] = fminf(m0[i], d0[i]);
                m1[i] = fminf(m1[i], d1[i]);
            }
        }

        if (have_next) {
            wait_async_done();     // my DMA done; barrier covers everyone else's
            __syncthreads();
            cur ^= 1;
        }
    }

    // Min across the 16 lanes holding each row's columns (C/D layout halves).
#pragma unroll
    for (int mask = 1; mask <= 8; mask <<= 1) {
#pragma unroll
        for (int i = 0; i < 8; i++) {
            m0[i] = fminf(m0[i], __shfl_xor(m0[i], mask, 32));
            m1[i] = fminf(m1[i], __shfl_xor(m1[i], mask, 32));
        }
    }
    // Per-lane sum over its 8 rows, then combine the two lane-halves.
    float s = 0.f;
#pragma unroll
    for (int i = 0; i < 8; i++) s += m0[i] + m1[i];
    s += __shfl_xor(s, 16, 32);          // sum of this wave's 32 row-mins

    // Add sum of ||a||^2 over this wave's rows (count each row once: lanes 0-15).
    float n = lo ? (norm0 + norm1) : 0.f;
#pragma unroll
    for (int mask = 1; mask <= 16; mask <<= 1) n += __shfl_xor(n, mask, 32);

    if (lane == 0) wsum[wave] = s + n;
    __syncthreads();
    if (tid == 0) {
        float acc = 0.f;
#pragma unroll
        for (int w = 0; w < WAVES; w++) acc += wsum[w];
        partials[blockIdx.x] = acc;
    }
}

__global__ void __launch_bounds__(BLOCK)
final_reduce(const float* __restrict__ partials, float* __restrict__ out) {
    __shared__ float sh[BLOCK];
    int tid = threadIdx.x;
    float acc = 0.f;
    for (int i = tid; i < N_WGS; i += BLOCK) acc += partials[i];
    sh[tid] = acc;
    __syncthreads();
    for (int stride = BLOCK / 2; stride > 0; stride >>= 1) {
        if (tid < stride) sh[tid] += sh[tid + stride];
        __syncthreads();
    }
    if (tid == 0) out[0] = sh[0] * (1.0f / (float)N_OUT);
}

extern "C" void kernel_launch(void* const* d_in, const int* in_sizes, int n_in,
                              void* d_out, int out_size, void* d_ws, size_t ws_size,
                              hipStream_t stream) {
    const float* outputs = (const float*)d_in[0];   // [131072, 3] f32
    const float* targets = (const float*)d_in[1];   // [16384, 3]  f32

    float4* t4      = (float4*)d_ws;                                  // 256 KB
    float* partials = (float*)((char*)d_ws + M_TGT * sizeof(float4)); // 2 KB

    prep_targets<<<M_TGT / 256, 256, 0, stream>>>(targets, t4);
    cpl_main<<<N_WGS, BLOCK, 0, stream>>>(outputs, t4, partials);
    final_reduce<<<1, BLOCK, 0, stream>>>(partials, (float*)d_out);
}